// SoftAttention_14671608283383
// MI455X (gfx1250) — compile-verified
//
#include <hip/hip_runtime.h>
#include <hip/hip_bf16.h>
#include <math.h>

typedef __attribute__((ext_vector_type(2))) float v2f;
typedef __attribute__((ext_vector_type(4))) float v4f;
typedef __attribute__((ext_vector_type(8))) float v8f;

#define B_ 32
#define S_ 2048
#define D_ 1024
#define TS 128      // s-rows per block in scores kernel
#define DC 64       // d-chunk staged in LDS per iteration
#define LDS_STRIDE 68  // 64 + 4 pad: (4*m + k) bank pattern -> conflict-free A reads

// ---------------------------------------------------------------------------
// Kernel 1: masked scores[b,s] = dot(context[b,s,:], h[b,:]) via fp32 WMMA.
// Block = 128 threads (4 waves). Each wave owns two 16-row WMMA tiles.
// B-fragment = h chunk replicated across all 16 N-columns, so every column of
// the 16x16 C tile holds the same dot product.
// ---------------------------------------------------------------------------
__global__ __launch_bounds__(128) void scores_kernel(
    const float* __restrict__ h, const float* __restrict__ ctx,
    const int* __restrict__ mask, float* __restrict__ scores)
{
    __shared__ float tile[TS * LDS_STRIDE];
    __shared__ float hl[D_];

    const int b    = blockIdx.y;
    const int s0   = blockIdx.x * TS;
    const int t    = threadIdx.x;
    const int lane = t & 31;
    const int wave = t >> 5;
    const int l16  = lane & 15;
    const int hi   = lane >> 4;       // 0 for lanes 0-15, 1 for lanes 16-31
    const int koff = hi << 1;         // K offset {0,2} per A/B VGPR layout

    // Preload full h[b,:] (1024 floats) into LDS, coalesced float4.
    #pragma unroll
    for (int j = 0; j < 2; ++j) {
        int i = t + 128 * j;          // float4 index 0..255
        *(v4f*)&hl[i * 4] = *(const v4f*)(h + (size_t)b * D_ + i * 4);
    }

    v8f acc0 = {};
    v8f acc1 = {};

    for (int dc = 0; dc < D_ / DC; ++dc) {
        __syncthreads();
        // Cooperative coalesced load: 128 rows x 64 floats -> LDS (padded).
        #pragma unroll
        for (int j = 0; j < 16; ++j) {
            int i  = t + 128 * j;     // float4 index 0..2047
            int r  = i >> 4;          // row 0..127
            int c4 = i & 15;          // float4 column 0..15
            v4f v = *(const v4f*)(ctx + ((size_t)b * S_ + s0 + r) * (size_t)D_
                                      + dc * DC + c4 * 4);
            *(v4f*)&tile[r * LDS_STRIDE + c4 * 4] = v;
        }
        __syncthreads();

        const int r0 = (wave * 32 + l16) * LDS_STRIDE;
        const int r1 = (wave * 32 + 16 + l16) * LDS_STRIDE;
        #pragma unroll
        for (int k = 0; k < DC; k += 4) {
            // B fragment: h[dchunk + k .. k+3] replicated over all N columns.
            v2f bb = *(const v2f*)&hl[dc * DC + k + koff];
            v2f a0 = *(const v2f*)&tile[r0 + k + koff];
            acc0 = __builtin_amdgcn_wmma_f32_16x16x4_f32(
                       false, a0, false, bb, (short)0, acc0, false, false);
            v2f a1 = *(const v2f*)&tile[r1 + k + koff];
            acc1 = __builtin_amdgcn_wmma_f32_16x16x4_f32(
                       false, a1, false, bb, (short)0, acc1, false, false);
        }
    }

    // C layout: lane n<16 -> M=r (VGPR r), lane n>=16 -> M=8+r. All N columns
    // identical, so lanes 0 and 16 of each wave write 8 rows each.
    if (l16 == 0) {
        const int mbase = hi * 8;
        #pragma unroll
        for (int r = 0; r < 8; ++r) {
            int sA = s0 + wave * 32 + mbase + r;
            scores[(size_t)b * S_ + sA] =
                mask[(size_t)b * S_ + sA] ? acc0[r] : -__builtin_inff();
            int sB = s0 + wave * 32 + 16 + mbase + r;
            scores[(size_t)b * S_ + sB] =
                mask[(size_t)b * S_ + sB] ? acc1[r] : -__builtin_inff();
        }
    }
}

// ---------------------------------------------------------------------------
// Kernel 2: in-place masked softmax over S per batch. 32 blocks x 256 threads.
// ---------------------------------------------------------------------------
__global__ __launch_bounds__(256) void softmax_kernel(float* __restrict__ attn)
{
    __shared__ float red[256];
    const int b = blockIdx.x;
    const int t = threadIdx.x;

    float v[8];
    float mx = -__builtin_inff();
    #pragma unroll
    for (int j = 0; j < 8; ++j) {
        v[j] = attn[(size_t)b * S_ + t + 256 * j];
        mx = fmaxf(mx, v[j]);
    }
    red[t] = mx;
    __syncthreads();
    for (int o = 128; o > 0; o >>= 1) {
        if (t < o) red[t] = fmaxf(red[t], red[t + o]);
        __syncthreads();
    }
    mx = red[0];
    __syncthreads();

    float sum = 0.f;
    #pragma unroll
    for (int j = 0; j < 8; ++j) { v[j] = __expf(v[j] - mx); sum += v[j]; }
    red[t] = sum;
    __syncthreads();
    for (int o = 128; o > 0; o >>= 1) {
        if (t < o) red[t] += red[t + o];
        __syncthreads();
    }
    float inv = 1.0f / red[0];
    #pragma unroll
    for (int j = 0; j < 8; ++j)
        attn[(size_t)b * S_ + t + 256 * j] = v[j] * inv;
}

// ---------------------------------------------------------------------------
// Kernel 3: partial weighted sums. Block = 256 threads, each owns 4 d-values
// (full D=1024 per block). blockIdx.x = S-split segment, blockIdx.y = batch.
// Writes [split][b][d] partials (or d_out directly when nsplit==1).
// ---------------------------------------------------------------------------
__global__ __launch_bounds__(256) void wsum_kernel(
    const float* __restrict__ ctx, const float* __restrict__ attn,
    float* __restrict__ outp, int nsplit)
{
    __shared__ float al[S_];
    const int b      = blockIdx.y;
    const int split  = blockIdx.x;
    const int t      = threadIdx.x;
    const int seg    = S_ / nsplit;
    const int sstart = split * seg;

    for (int j = t; j < seg; j += 256)
        al[j] = attn[(size_t)b * S_ + sstart + j];
    __syncthreads();

    v4f acc = {};
    const float* base = ctx + ((size_t)b * S_ + sstart) * (size_t)D_ + 4 * t;
    #pragma unroll 4
    for (int s = 0; s < seg; ++s) {
        float a = al[s];
        v4f cv = *(const v4f*)(base + (size_t)s * D_);
        acc += a * cv;
    }
    *(v4f*)&outp[((size_t)split * B_ + b) * (size_t)D_ + 4 * t] = acc;
}

// ---------------------------------------------------------------------------
// Kernel 4: deterministic reduction of S-split partials into d_out.
// ---------------------------------------------------------------------------
__global__ __launch_bounds__(256) void reduce_kernel(
    const float* __restrict__ part, float* __restrict__ outp, int nsplit)
{
    int i = blockIdx.x * 256 + threadIdx.x;   // over B*D
    float s = 0.f;
    for (int k = 0; k < nsplit; ++k)
        s += part[(size_t)k * B_ * D_ + i];
    outp[i] = s;
}

extern "C" void kernel_launch(void* const* d_in, const int* in_sizes, int n_in,
                              void* d_out, int out_size, void* d_ws, size_t ws_size,
                              hipStream_t stream)
{
    const float* h    = (const float*)d_in[0];   // [B, D]
    const float* ctx  = (const float*)d_in[1];   // [B, S, D]
    const int*   mask = (const int*)d_in[2];     // [B, S]

    float* wc   = (float*)d_out;                 // [B, D]   (output 0)
    float* attn = wc + (size_t)B_ * D_;          // [B, S]   (output 1)

    // Pass 1: masked scores into the attn output region (WMMA GEMV).
    scores_kernel<<<dim3(S_ / TS, B_), 128, 0, stream>>>(h, ctx, mask, attn);

    // Pass 2: in-place softmax.
    softmax_kernel<<<dim3(B_), 256, 0, stream>>>(attn);

    // Pass 3: weighted context. Split S 8 ways if scratch allows (more waves
    // in flight to saturate HBM), else single pass straight into d_out.
    const int nsplit = (ws_size >= (size_t)8 * B_ * D_ * sizeof(float)) ? 8 : 1;
    if (nsplit > 1) {
        wsum_kernel<<<dim3(nsplit, B_), 256, 0, stream>>>(ctx, attn,
                                                          (float*)d_ws, nsplit);
        reduce_kernel<<<dim3((B_ * D_) / 256), 256, 0, stream>>>(
            (const float*)d_ws, wc, nsplit);
    } else {
        wsum_kernel<<<dim3(1, B_), 256, 0, stream>>>(ctx, attn, wc, 1);
    }
}